// FilterInitializerSiamese_59081570124310
// MI455X (gfx1250) — compile-verified
//
#include <hip/hip_runtime.h>
#include <hip/hip_bf16.h>

// WMMA operand types (CDNA5, wave32)
typedef __attribute__((ext_vector_type(2))) float v2f;
typedef __attribute__((ext_vector_type(8))) float v8f;

// LDS layout: B matrix 16 rows (pq) x 486 dwords pitch (484 used, pitch chosen so
// lane-strided ds_load_b64 is bank-conflict free: 486 mod 64 = 38, 19*d != 0 mod 32
// for d=1..15, and cross-half offset +2 can never collide), then Yh (4x22) and Xw (4x22).
#define BPITCH     486
#define YH_OFF     (16 * BPITCH)
#define XW_OFF     (16 * BPITCH + 88)
#define LDS_FLOATS (16 * BPITCH + 176)

__device__ __forceinline__ float hat_cdf(float t) {
    // Antiderivative of hat(x)=max(0,1-|x|), branchless:
    // t<=-1 -> 0 ; (-1,0] -> 0.5(t+1)^2 ; (0,1] -> 0.5+t-0.5t^2 ; >1 -> 1
    float a = fminf(fmaxf(t + 1.0f, 0.0f), 1.0f);
    float b = fminf(fmaxf(t, 0.0f), 1.0f);
    return 0.5f * a * a + b - 0.5f * b * b;
}

__global__ __launch_bounds__(256)
void prpool_wmma_kernel(const float* __restrict__ feat,
                        const float* __restrict__ bb,
                        float* __restrict__ out) {
    __shared__ float lds[LDS_FLOATS];

    const int tid    = threadIdx.x;
    const int lane   = tid & 31;
    const int wave   = tid >> 5;
    const int n      = lane & 15;        // N index (pq) / M index (row in tile)
    const int koff   = (lane >> 4) * 2;  // K sub-offset per ISA A/B layout
    const int s      = blockIdx.y;       // sequence 0..63
    const int cgroup = blockIdx.x;       // 0..3
    const int ctbase = cgroup * 128 + wave * 16;  // first channel of this wave's tile

    v8f acc = {};  // 16x16 f32 accumulator, summed over the 3 images

    for (int i = 0; i < 3; ++i) {
        const int roi = i * 64 + s;

        // ---- phase 1: 1D exact hat integrals Yh(4x22), Xw(4x22) into LDS ----
        if (tid < 176) {
            const float x1    = bb[roi * 4 + 0] * (1.0f / 16.0f);
            const float y1    = bb[roi * 4 + 1] * (1.0f / 16.0f);
            const float bw    = bb[roi * 4 + 2] * (1.0f / 16.0f) * 0.25f;  // bin_w
            const float bh    = bb[roi * 4 + 3] * (1.0f / 16.0f) * 0.25f;  // bin_h
            const float area  = fmaxf(bw * bh, 0.0f);
            // fold 1/area, mean over 3 images, and FILTER_NORM 1/(C*P*P)=1/8192
            const float inv   = (area > 0.0f) ? (1.0f / (area * 3.0f * 8192.0f)) : 0.0f;
            const int   r     = tid / 22;
            const int   j     = tid - r * 22;
            const float jf    = (float)j;
            if (r < 4) {
                const float sy = y1 + (float)r * bh;
                lds[YH_OFF + r * 22 + j] = hat_cdf(sy + bh - jf) - hat_cdf(sy - jf);
            } else {
                const int   q  = r - 4;
                const float sx = x1 + (float)q * bw;
                lds[XW_OFF + q * 22 + j] = (hat_cdf(sx + bw - jf) - hat_cdf(sx - jf)) * inv;
            }
        }
        __syncthreads();

        // ---- phase 2: outer-product B[pq][hw] = Yh[p][h] * Xw[q][w] (shared by 8 waves) ----
        for (int idx = tid; idx < 16 * 484; idx += 256) {
            const int pq = idx / 484;
            const int hw = idx - pq * 484;
            const int h  = hw / 22;
            const int w  = hw - h * 22;
            lds[pq * BPITCH + hw] =
                lds[YH_OFF + (pq >> 2) * 22 + h] * lds[XW_OFF + (pq & 3) * 22 + w];
        }
        __syncthreads();

        // ---- phase 3: D += A(16x4 feat rows) x B(4x16 kernel), K = 484 in steps of 4 ----
        // A layout: lane L -> M = L&15, VGPR{0,1} = K = kb + 2*(L>>4) + {0,1}
        const float* ap = feat + ((size_t)(roi * 512 + ctbase + n)) * 484 + koff;
        const float* bl = &lds[n * BPITCH + koff];
        #pragma unroll
        for (int kb = 0; kb < 484; kb += 4) {
            v2f a = *(const v2f*)(ap + kb);  // global_load_b64, immediate offset
            v2f b = *(const v2f*)(bl + kb);  // ds_load_b64, conflict-free
            acc = __builtin_amdgcn_wmma_f32_16x16x4_f32(
                /*neg_a=*/false, a, /*neg_b=*/false, b,
                /*c_mod=*/(short)0, acc, /*reuse_a=*/false, /*reuse_b=*/false);
        }
        __syncthreads();  // protect B/Yh/Xw before next image overwrites them
    }

    // ---- store: C/D layout: lane L, VGPR r -> M = r + 8*(L>>4), N = L&15 ----
    #pragma unroll
    for (int r = 0; r < 8; ++r) {
        const int c = ctbase + (lane >> 4) * 8 + r;
        out[((size_t)s * 512 + c) * 16 + n] = acc[r];
    }
}

extern "C" void kernel_launch(void* const* d_in, const int* in_sizes, int n_in,
                              void* d_out, int out_size, void* d_ws, size_t ws_size,
                              hipStream_t stream) {
    const float* feat = (const float*)d_in[0];  // (3, 64, 512, 22, 22) fp32
    const float* bb   = (const float*)d_in[1];  // (3, 64, 4) fp32
    float* out        = (float*)d_out;          // (64, 512, 4, 4) fp32

    dim3 grid(4, 64);   // (c-group, seq) -> 256 blocks x 8 waves = 2048 WMMA waves
    prpool_wmma_kernel<<<grid, 256, 0, stream>>>(feat, bb, out);
}